// LunarLanderEKF_26061861552526
// MI455X (gfx1250) — compile-verified
//
#include <hip/hip_runtime.h>
#include <stdint.h>

// ---------------------------------------------------------------------------
// LunarLander EKF, batch = 524288 independent 6-state filters.
// Memory-bound (~187 MB traffic -> ~8 us at 23.3 TB/s). One element/thread.
// CDNA5 path used: async global<->LDS staging (ASYNCcnt) for the 144B-strided
// P matrices, so every VMEM instruction is fully coalesced (512 B / wave).
// ---------------------------------------------------------------------------

#define TPB 256
#define EKF_DT 0.02f

// ---- CDNA5 async global<->LDS ops (gfx1250), via inline asm --------------
__device__ __forceinline__ void async_load_b128(unsigned lds_addr, const void* gaddr) {
  // copies 16 bytes from global address gaddr to LDS address lds_addr (per lane, ASYNCcnt)
  asm volatile("global_load_async_to_lds_b128 %0, %1, off"
               :: "v"(lds_addr), "v"(gaddr) : "memory");
}
__device__ __forceinline__ void async_store_b128(void* gaddr, unsigned lds_addr) {
  // copies 16 bytes from LDS address lds_addr to global address gaddr (per lane, ASYNCcnt)
  asm volatile("global_store_async_from_lds_b128 %0, %1, off"
               :: "v"(gaddr), "v"(lds_addr) : "memory");
}
__device__ __forceinline__ void wait_async0() {
  asm volatile("s_wait_asynccnt 0" ::: "memory");
}
__device__ __forceinline__ unsigned lds_byte_addr(const void* p) {
  // flat->LDS aperture: LDS byte offset is the low 32 bits of a generic
  // pointer to __shared__ memory (ISA 10.2).
  return (unsigned)(uintptr_t)p;
}

// ---- prelude: build uniform Q (6x6) and R (3x3) into ws[0..44] -----------
__global__ void ekf_build_qr(const float* __restrict__ qld, const float* __restrict__ qod,
                             const float* __restrict__ rld, const float* __restrict__ rod,
                             float* __restrict__ ws) {
  if (threadIdx.x != 0 || blockIdx.x != 0) return;
  float L[6][6];
  for (int i = 0; i < 6; ++i)
    for (int j = 0; j < 6; ++j) L[i][j] = 0.0f;
  for (int i = 0; i < 6; ++i) L[i][i] = expf(qld[i]);
  int k = 0;
  for (int r = 1; r < 6; ++r)          // np.tril_indices(6,-1) order
    for (int c = 0; c < r; ++c) L[r][c] = qod[k++];
  for (int i = 0; i < 6; ++i)
    for (int j = 0; j < 6; ++j) {
      float s = 0.0f;
      for (int t = 0; t < 6; ++t) s += L[i][t] * L[j][t];
      if (i == j) s += 1e-6f;
      ws[i * 6 + j] = s;
    }
  float M[3][3];
  for (int i = 0; i < 3; ++i)
    for (int j = 0; j < 3; ++j) M[i][j] = 0.0f;
  for (int i = 0; i < 3; ++i) M[i][i] = expf(rld[i]);
  k = 0;
  for (int r = 1; r < 3; ++r)
    for (int c = 0; c < r; ++c) M[r][c] = rod[k++];
  for (int i = 0; i < 3; ++i)
    for (int j = 0; j < 3; ++j) {
      float s = 0.0f;
      for (int t = 0; t < 3; ++t) s += M[i][t] * M[j][t];
      if (i == j) s += 1e-6f;
      ws[36 + i * 3 + j] = s;
    }
}

// ---- main EKF kernel ------------------------------------------------------
__global__ __launch_bounds__(TPB) void ekf_main(
    const float* __restrict__ z, const float* __restrict__ u,
    const float* __restrict__ xprev, const float* __restrict__ Pprev,
    const float* __restrict__ qr,
    float* __restrict__ xout, float* __restrict__ pout)
{
  __shared__ __align__(16) float sP[TPB * 36];   // staged P_prev tile (36 KB)
  __shared__ __align__(16) float sO[TPB * 36];   // staged P_upd tile (36 KB)

  const int t = threadIdx.x;
  const long long e0 = (long long)blockIdx.x * TPB;   // first element of block
  const float dt = EKF_DT;

  // -- stage this block's P_prev (TPB*144 B) into LDS, fully coalesced -----
  {
    const char* gsrc = (const char*)(Pprev + e0 * 36);
    const unsigned lbase = lds_byte_addr(&sP[0]);
#pragma unroll
    for (int k = 0; k < 9; ++k) {
      const unsigned off = (unsigned)((t + k * TPB) * 16);
      async_load_b128(lbase + off, gsrc + off);
    }
  }

  // -- small per-element inputs: direct (contiguous-enough) loads ----------
  const float z0 = z[(e0 + t) * 3 + 0];
  const float z1 = z[(e0 + t) * 3 + 1];
  const float z2 = z[(e0 + t) * 3 + 2];
  const float2 uv = *(const float2*)(u + (e0 + t) * 2);
  const float2* xv = (const float2*)(xprev + (e0 + t) * 6);
  const float2 x01 = xv[0], x23 = xv[1], x45 = xv[2];
  const float px = x01.x, py = x01.y, vx = x23.x, vy = x23.y, th = x45.x, om = x45.y;

  // -- dynamics + Jacobian scalars -----------------------------------------
  const float mp = fminf(fmaxf(uv.x, 0.0f), 1.0f);
  float s, c;
  __sincosf(th, &s, &c);
  const float Tm = 6.0f * mp;                  // MAIN_THRUST * main_power
  const float nvx = vx + (-Tm * s) * dt;
  const float nvy = vy + (Tm * c - 10.0f) * dt;
  const float nom = om + uv.y * dt;            // SIDE_TORQUE = 1
  const float npx = px + nvx * dt;
  const float npy = py + nvy * dt;
  const float nth = th + nom * dt;
  float xpred[6] = {npx, npy, nvx, nvy, nth, nom};
  const float fa = -Tm * c * dt;               // F[2][4]
  const float fb = -Tm * s * dt;               // F[3][4]  (F[0][4]=dt*fa, F[1][4]=dt*fb)

  // -- wait for staged tile, then pull own 144 B from LDS ------------------
  wait_async0();
  __syncthreads();

  float P[36];
  {
    const float4* src = (const float4*)&sP[t * 36];
#pragma unroll
    for (int k = 0; k < 9; ++k) {
      const float4 v = src[k];
      P[4 * k + 0] = v.x; P[4 * k + 1] = v.y; P[4 * k + 2] = v.z; P[4 * k + 3] = v.w;
    }
  }
#define PP(i, j) P[(i) * 6 + (j)]

  // -- P_pred = F P F^T + Q  (F = I + sparse; two in-place sweeps) ---------
#pragma unroll
  for (int j = 0; j < 6; ++j) {                // rows: M = F P
    const float p4 = PP(4, j);
    PP(0, j) = PP(0, j) + dt * PP(2, j) + (dt * fa) * p4;
    PP(1, j) = PP(1, j) + dt * PP(3, j) + (dt * fb) * p4;
    PP(2, j) = PP(2, j) + fa * p4;
    PP(3, j) = PP(3, j) + fb * p4;
    PP(4, j) = p4 + dt * PP(5, j);
  }
#pragma unroll
  for (int i = 0; i < 6; ++i) {                // cols: (M) F^T
    const float m4 = PP(i, 4);
    PP(i, 0) = PP(i, 0) + dt * PP(i, 2) + (dt * fa) * m4;
    PP(i, 1) = PP(i, 1) + dt * PP(i, 3) + (dt * fb) * m4;
    PP(i, 2) = PP(i, 2) + fa * m4;
    PP(i, 3) = PP(i, 3) + fb * m4;
    PP(i, 4) = m4 + dt * PP(i, 5);
  }
#pragma unroll
  for (int i = 0; i < 36; ++i) P[i] += qr[i];  // + Q (uniform)

  // -- innovation, S = HP H^T + R (H selects rows/cols {0,1,4}) ------------
  const float y0 = z0 - npx, y1 = z1 - npy, y2 = z2 - nth;
  const float* Rm = qr + 36;
  const float S00 = PP(0, 0) + Rm[0], S01 = PP(0, 1) + Rm[1], S02 = PP(0, 4) + Rm[2];
  const float S10 = PP(1, 0) + Rm[3], S11 = PP(1, 1) + Rm[4], S12 = PP(1, 4) + Rm[5];
  const float S20 = PP(4, 0) + Rm[6], S21 = PP(4, 1) + Rm[7], S22 = PP(4, 4) + Rm[8];
  const float c00 = S11 * S22 - S12 * S21;
  const float c01 = S12 * S20 - S10 * S22;
  const float c02 = S10 * S21 - S11 * S20;
  const float inv = 1.0f / (S00 * c00 + S01 * c01 + S02 * c02);
  const float i00 = c00 * inv;
  const float i01 = (S02 * S21 - S01 * S22) * inv;
  const float i02 = (S01 * S12 - S02 * S11) * inv;
  const float i10 = c01 * inv;
  const float i11 = (S00 * S22 - S02 * S20) * inv;
  const float i12 = (S02 * S10 - S00 * S12) * inv;
  const float i20 = c02 * inv;
  const float i21 = (S01 * S20 - S00 * S21) * inv;
  const float i22 = (S00 * S11 - S01 * S10) * inv;

  // -- K = (S^{-1} HP)^T : K[j][m] = sum_n Sinv[m][n] * Ppred[r_n][j] ------
  float K[6][3];
#pragma unroll
  for (int j = 0; j < 6; ++j) {
    const float h0 = PP(0, j), h1 = PP(1, j), h2 = PP(4, j);
    K[j][0] = i00 * h0 + i01 * h1 + i02 * h2;
    K[j][1] = i10 * h0 + i11 * h1 + i12 * h2;
    K[j][2] = i20 * h0 + i21 * h1 + i22 * h2;
  }

  // -- x_upd ---------------------------------------------------------------
  float xo[6];
#pragma unroll
  for (int j = 0; j < 6; ++j)
    xo[j] = xpred[j] + K[j][0] * y0 + K[j][1] * y1 + K[j][2] * y2;

  // -- Joseph form: P_upd = (I-KH) Ppred (I-KH)^T + K R K^T ----------------
  float r0[6], r1[6], r4[6];
#pragma unroll
  for (int j = 0; j < 6; ++j) { r0[j] = PP(0, j); r1[j] = PP(1, j); r4[j] = PP(4, j); }
#pragma unroll
  for (int i = 0; i < 6; ++i) {                // N = (I-KH) Ppred, in place
    const float k0 = K[i][0], k1 = K[i][1], k2 = K[i][2];
#pragma unroll
    for (int j = 0; j < 6; ++j)
      PP(i, j) = PP(i, j) - k0 * r0[j] - k1 * r1[j] - k2 * r4[j];
  }
#pragma unroll
  for (int i = 0; i < 6; ++i) {                // N (I-KH)^T, in place
    const float n0 = PP(i, 0), n1 = PP(i, 1), n4 = PP(i, 4);
#pragma unroll
    for (int j = 0; j < 6; ++j)
      PP(i, j) = PP(i, j) - K[j][0] * n0 - K[j][1] * n1 - K[j][2] * n4;
  }
#pragma unroll
  for (int i = 0; i < 6; ++i) {                // + K R K^T
    const float t0 = K[i][0] * Rm[0] + K[i][1] * Rm[3] + K[i][2] * Rm[6];
    const float t1 = K[i][0] * Rm[1] + K[i][1] * Rm[4] + K[i][2] * Rm[7];
    const float t2 = K[i][0] * Rm[2] + K[i][1] * Rm[5] + K[i][2] * Rm[8];
#pragma unroll
    for (int j = 0; j < 6; ++j)
      PP(i, j) += t0 * K[j][0] + t1 * K[j][1] + t2 * K[j][2];
  }

  // -- outputs: x_upd direct; P_upd staged through LDS + async store -------
  {
    float2* xov = (float2*)(xout + (e0 + t) * 6);
    xov[0] = make_float2(xo[0], xo[1]);
    xov[1] = make_float2(xo[2], xo[3]);
    xov[2] = make_float2(xo[4], xo[5]);
  }
  {
    float4* dst = (float4*)&sO[t * 36];
#pragma unroll
    for (int k = 0; k < 9; ++k)
      dst[k] = make_float4(P[4 * k + 0], P[4 * k + 1], P[4 * k + 2], P[4 * k + 3]);
  }
  __syncthreads();                             // all P_upd tiles in LDS
  {
    char* gdst = (char*)(pout + e0 * 36);
    const unsigned lbase = lds_byte_addr(&sO[0]);
#pragma unroll
    for (int k = 0; k < 9; ++k) {
      const unsigned off = (unsigned)((t + k * TPB) * 16);
      async_store_b128(gdst + off, lbase + off);
    }
  }
  wait_async0();                               // retire async stores
#undef PP
}

extern "C" void kernel_launch(void* const* d_in, const int* in_sizes, int n_in,
                              void* d_out, int out_size, void* d_ws, size_t ws_size,
                              hipStream_t stream)
{
  const float* z   = (const float*)d_in[0];
  const float* u   = (const float*)d_in[1];
  const float* xp  = (const float*)d_in[2];
  const float* Pp  = (const float*)d_in[3];
  const float* qld = (const float*)d_in[4];
  const float* qod = (const float*)d_in[5];
  const float* rld = (const float*)d_in[6];
  const float* rod = (const float*)d_in[7];
  const int B = in_sizes[0] / 3;               // z is (B,3)
  float* ws   = (float*)d_ws;                  // Q(36) + R(9)
  float* xout = (float*)d_out;                 // (B,6)
  float* pout = (float*)d_out + (size_t)B * 6; // (B,6,6)

  ekf_build_qr<<<1, 64, 0, stream>>>(qld, qod, rld, rod, ws);
  ekf_main<<<B / TPB, TPB, 0, stream>>>(z, u, xp, Pp, ws, xout, pout);
}